// PaG_14070312861866
// MI455X (gfx1250) — compile-verified
//
#include <hip/hip_runtime.h>
#include <math.h>

// ---------------------------------------------------------------------------
// Shapes
// ---------------------------------------------------------------------------
#define BB 8
#define SS 256
#define DD 300
#define EE 200
#define PD 100          // POSI_DIM
#define RN 12           // REL_NUM
#define CAH 768
#define ROWS (BB * SS)  // 2048
#define KCAT (13 * DD)  // 3900 (12 relation blocks + root block)

typedef float v2f __attribute__((ext_vector_type(2)));
typedef float v4f __attribute__((ext_vector_type(4)));
typedef float v8f __attribute__((ext_vector_type(8)));

static __device__ __forceinline__ int imin(int a, int b) { return a < b ? a : b; }
static __device__ __forceinline__ int imax(int a, int b) { return a > b ? a : b; }

// ---------------------------------------------------------------------------
// 1) rel_emb_k / rel_emb_v broadcast:  out[b,i,j,:] = pe[clip(i-j+1,0,256)]
//    2 x 210 MB of write-once streaming -> nontemporal 128-bit stores so the
//    stream does not evict the GEMM working set from L2. pe tables (~103 KB)
//    stay cached.
// ---------------------------------------------------------------------------
__global__ __launch_bounds__(256) void rel_emb_kernel(
    const float* __restrict__ pe_k, const float* __restrict__ pe_v,
    float* __restrict__ out_k, float* __restrict__ out_v) {
  const int bi = blockIdx.x;          // 0..2047
  const int i  = bi & (SS - 1);
  const v4f* pk = (const v4f*)pe_k;   // rows of 25 v4f (100 floats)
  const v4f* pv = (const v4f*)pe_v;
  v4f* ok = (v4f*)out_k + (long)bi * SS * (PD / 4);
  v4f* ov = (v4f*)out_v + (long)bi * SS * (PD / 4);
  for (int vi = threadIdx.x; vi < SS * (PD / 4); vi += 256) {
    const int j = vi / (PD / 4);
    const int c = vi - j * (PD / 4);
    int pos = i - j + 1;
    pos = pos < 0 ? 0 : (pos > SS ? SS : pos);
    __builtin_nontemporal_store(pk[pos * (PD / 4) + c], &ok[vi]);
    __builtin_nontemporal_store(pv[pos * (PD / 4) + c], &ov[vi]);
  }
}

// ---------------------------------------------------------------------------
// 2) Inclusive prefix sum over sequence index:  P[b,i,d] = sum_{i'<=i} x[b,i',d]
// ---------------------------------------------------------------------------
__global__ __launch_bounds__(320) void prefix_kernel(const float* __restrict__ x,
                                                     float* __restrict__ P) {
  const int b = blockIdx.x;
  const int d = threadIdx.x;
  if (d >= DD) return;
  const float* xb = x + (long)b * SS * DD;
  float* Pb = P + (long)b * SS * DD;
  float run = 0.f;
  for (int i = 0; i < SS; ++i) {
    run += xb[i * DD + d];
    Pb[i * DD + d] = run;
  }
}

// ---------------------------------------------------------------------------
// 3) Build A_cat [2048, 3900]: 12 per-relation means + root block (= x).
//    etype depends only on (i - j), so group sums come from prefix sums.
// ---------------------------------------------------------------------------
__global__ __launch_bounds__(320) void build_acat_kernel(
    const float* __restrict__ x, const float* __restrict__ P,
    float* __restrict__ A_cat) {
  const int bj = blockIdx.x;          // 0..2047
  const int b = bj >> 8;
  const int j = bj & (SS - 1);
  const int d = threadIdx.x;
  if (d >= DD) return;
  const float* Pb = P + (long)b * SS * DD;
  const float* xb = x + (long)b * SS * DD;
  float* row = A_cat + (long)bj * KCAT;

  const float xv = xb[j * DD + d];
  // r = 0 (self)
  row[0 * DD + d] = xv;
  // r = 1 (all i<j, plus far tail i>=j+21)
  {
    float s = (j >= 1) ? Pb[(j - 1) * DD + d] : 0.f;
    int cnt = j;
    if (j + 21 <= SS - 1) {
      s += Pb[(SS - 1) * DD + d] - Pb[(j + 20) * DD + d];
      cnt += (SS - 1) - (j + 21) + 1;
    }
    row[1 * DD + d] = s / (float)imax(cnt, 1);
  }
  // r = 2..11 : m = 12-r, sources i = j+2m-1, j+2m (clipped)
  for (int r = 2; r <= 11; ++r) {
    const int m = 12 - r;
    const int a = j + 2 * m - 1;
    const int e = j + 2 * m;
    float s = 0.f;
    int cnt = 0;
    if (a <= SS - 1) {
      const int ee = imin(e, SS - 1);
      s = Pb[ee * DD + d] - Pb[(a - 1) * DD + d];
      cnt = ee - a + 1;
    }
    row[r * DD + d] = s / (float)imax(cnt, 1);
  }
  // root block (r = 12): raw x, multiplied by rgcn_root in W_cat
  row[12 * DD + d] = xv;
}

// ---------------------------------------------------------------------------
// 4) Build W_catT [300, 3900] (TRANSPOSED, N-major):
//    W_catT[n, r*300+k] = (comp@basis)[r,k,n] for r<12;  root[k,n] for r=12.
//    K-major rows let the GEMM read B fragments as aligned float2.
// ---------------------------------------------------------------------------
__global__ __launch_bounds__(256) void build_wcatT_kernel(
    const float* __restrict__ comp, const float* __restrict__ basis,
    const float* __restrict__ root, float* __restrict__ W) {
  const long idx = (long)blockIdx.x * blockDim.x + threadIdx.x;
  if (idx >= (long)DD * KCAT) return;
  const int n = (int)(idx / KCAT);
  const int kk = (int)(idx % KCAT);
  const int r = kk / DD;
  const int k = kk - r * DD;
  if (r < RN) {
    float s = 0.f;
#pragma unroll
    for (int bb = 0; bb < 4; ++bb)
      s += comp[r * 4 + bb] * basis[((long)bb * DD + k) * DD + n];
    W[idx] = s;
  } else {
    W[idx] = root[(long)k * DD + n];
  }
}

// ---------------------------------------------------------------------------
// 5) Generic transpose: dst[N,K] = src[K,N]^T, batched via grid.y.
// ---------------------------------------------------------------------------
__global__ __launch_bounds__(256) void transpose_kernel(
    const float* __restrict__ src, float* __restrict__ dst, int K, int N,
    long sS, long sD) {
  const int z = blockIdx.y;
  src += (long)z * sS;
  dst += (long)z * sD;
  const long idx = (long)blockIdx.x * blockDim.x + threadIdx.x;
  if (idx >= (long)K * N) return;
  const int n = (int)(idx / K);
  const int k = (int)(idx % K);
  dst[idx] = src[(long)k * N + n];
}

// ---------------------------------------------------------------------------
// 6) fp32 WMMA GEMM:  C = act(A[M,K] * B^T + bias),  B given as [N,K] rows.
//    Both operands are K-major -> K-pair fragments load as aligned float2
//    (global_load_b64). Block = 256 threads = 8 waves, 4(M) x 2(N); each
//    wave owns a 32x64 tile = 2x4 fragments of V_WMMA_F32_16X16X4_F32
//    (8 WMMA per 6 b64 loads in the K loop). Batched via grid.z.
//    A frag (16x4 f32):  lane = M%16 + 16*(K/2), vgpr = K%2  (ISA 7.12.2)
//    B frag (4x16 f32):  lane = N%16 + 16*(K/2), vgpr = K%2
//    C/D  (16x16 f32):   vgpr v -> M = v + 8*lane[4], N = lane%16
// ---------------------------------------------------------------------------
template <int ACT>
__global__ __launch_bounds__(256) void gemm_f32_wmma(
    const float* __restrict__ A, long sA, int lda,
    const float* __restrict__ B, long sB, int ldb,
    float* __restrict__ C, long sC, int ldc,
    int M, int N, int K, const float* __restrict__ bias) {
  const int z = blockIdx.z;
  A += (long)z * sA;
  B += (long)z * sB;
  C += (long)z * sC;

  const int lane = threadIdx.x & 31;
  const int wave = threadIdx.x >> 5;
  const int mBase = blockIdx.y * 128 + (wave >> 1) * 32;
  const int nBase = blockIdx.x * 128 + (wave & 1) * 64;
  const int lo = lane & 15;
  const int hi = lane >> 4;

  const long am0 = (long)imin(mBase + lo, M - 1) * lda;
  const long am1 = (long)imin(mBase + 16 + lo, M - 1) * lda;
  long bnl[4];
#pragma unroll
  for (int t = 0; t < 4; ++t)
    bnl[t] = (long)imin(nBase + 16 * t + lo, N - 1) * ldb;

  v8f acc0[4] = {{}, {}, {}, {}};
  v8f acc1[4] = {{}, {}, {}, {}};

  for (int k0 = 0; k0 < K; k0 += 4) {
    if ((k0 & 63) == 0 && k0 + 64 < K) {
      __builtin_prefetch(A + am0 + k0 + 64, 0, 1);
      __builtin_prefetch(B + bnl[0] + k0 + 64, 0, 1);
    }
    const int ka = k0 + 2 * hi;  // this lane's K pair
    const v2f a0 = *(const v2f*)(A + am0 + ka);
    const v2f a1 = *(const v2f*)(A + am1 + ka);
    v2f bf[4];
#pragma unroll
    for (int t = 0; t < 4; ++t) bf[t] = *(const v2f*)(B + bnl[t] + ka);
#pragma unroll
    for (int t = 0; t < 4; ++t) {
      acc0[t] = __builtin_amdgcn_wmma_f32_16x16x4_f32(
          false, a0, false, bf[t], (short)0, acc0[t], false, false);
      acc1[t] = __builtin_amdgcn_wmma_f32_16x16x4_f32(
          false, a1, false, bf[t], (short)0, acc1[t], false, false);
    }
  }

  auto store_tile = [&](const v8f& acc, int tm, int tn) {
    const int n = tn + lo;
    if (n >= N) return;
    const float bv = bias ? bias[n] : 0.f;
#pragma unroll
    for (int v = 0; v < 8; ++v) {
      const int m = tm + v + 8 * hi;
      if (m < M) {
        float val = acc[v] + bv;
        if (ACT == 1) val = fmaxf(val, 0.f);
        C[(long)m * ldc + n] = val;
      }
    }
  };
#pragma unroll
  for (int t = 0; t < 4; ++t) {
    store_tile(acc0[t], mBase, nBase + 16 * t);
    store_tile(acc1[t], mBase + 16, nBase + 16 * t);
  }
}

// ---------------------------------------------------------------------------
// 7) Column softmax over axis=1 (s index) of scores[b, s, t], in place.
// ---------------------------------------------------------------------------
__global__ __launch_bounds__(256) void softmax_col_kernel(float* __restrict__ scores) {
  const int t = blockIdx.x;
  const int b = blockIdx.y;
  const int s = threadIdx.x;
  float* sc = scores + (long)b * SS * SS;
  __shared__ float red[SS];
  const float v = sc[(long)s * SS + t];
  red[s] = v;
  __syncthreads();
  for (int off = 128; off > 0; off >>= 1) {
    if (s < off) red[s] = fmaxf(red[s], red[s + off]);
    __syncthreads();
  }
  const float mx = red[0];
  __syncthreads();
  const float e = expf(v - mx);
  red[s] = e;
  __syncthreads();
  for (int off = 128; off > 0; off >>= 1) {
    if (s < off) red[s] += red[s + off];
    __syncthreads();
  }
  sc[(long)s * SS + t] = e / red[0];
}

// ---------------------------------------------------------------------------
// 8) DualViewGate: g = sigmoid(x . gw + gb);  out = x1*g1 + x2*g2
// ---------------------------------------------------------------------------
__global__ __launch_bounds__(256) void gate_kernel(
    const float* __restrict__ x1, const float* __restrict__ x2,
    const float* __restrict__ g1w, const float* __restrict__ g1b,
    const float* __restrict__ g2w, const float* __restrict__ g2b,
    float* __restrict__ out) {
  const long row = blockIdx.x;
  const int t = threadIdx.x;
  __shared__ float s1[256], s2[256];
  __shared__ float g1s, g2s;
  float p1 = 0.f, p2 = 0.f;
  for (int d = t; d < DD; d += 256) {
    p1 += x1[row * DD + d] * g1w[d];
    p2 += x2[row * DD + d] * g2w[d];
  }
  s1[t] = p1;
  s2[t] = p2;
  __syncthreads();
  for (int off = 128; off > 0; off >>= 1) {
    if (t < off) {
      s1[t] += s1[t + off];
      s2[t] += s2[t + off];
    }
    __syncthreads();
  }
  if (t == 0) {
    g1s = 1.f / (1.f + expf(-(s1[0] + g1b[0])));
    g2s = 1.f / (1.f + expf(-(s2[0] + g2b[0])));
  }
  __syncthreads();
  for (int d = t; d < DD; d += 256) {
    out[row * DD + d] = x1[row * DD + d] * g1s + x2[row * DD + d] * g2s;
  }
}

// ---------------------------------------------------------------------------
// Host launcher
// ---------------------------------------------------------------------------
extern "C" void kernel_launch(void* const* d_in, const int* in_sizes, int n_in,
                              void* d_out, int out_size, void* d_ws, size_t ws_size,
                              hipStream_t stream) {
  const float* x       = (const float*)d_in[0];   // [8,256,300]
  const float* emo     = (const float*)d_in[1];   // [8,256,200]
  const float* tf      = (const float*)d_in[2];   // [8,256,300]
  const float* pe_k    = (const float*)d_in[3];   // [257,100]
  const float* pe_v    = (const float*)d_in[4];   // [257,100]
  const float* comp    = (const float*)d_in[5];   // [12,4]
  const float* basis   = (const float*)d_in[6];   // [4,300,300]
  const float* root    = (const float*)d_in[7];   // [300,300]
  const float* rg_bias = (const float*)d_in[8];   // [300]
  const float* ca_wa   = (const float*)d_in[9];   // [200,768]
  const float* ca_ba   = (const float*)d_in[10];  // [768]
  const float* ca_wb   = (const float*)d_in[11];  // [300,768]
  const float* ca_bb   = (const float*)d_in[12];  // [768]
  const float* dv_w1   = (const float*)d_in[13];  // [300,300]
  const float* dv_b1   = (const float*)d_in[14];  // [300]
  const float* dv_w2   = (const float*)d_in[15];  // [300,300]
  const float* dv_b2   = (const float*)d_in[16];  // [300]
  const float* dv_g1w  = (const float*)d_in[17];  // [300,1]
  const float* dv_g1b  = (const float*)d_in[18];  // [1]
  const float* dv_g2w  = (const float*)d_in[19];  // [300,1]
  const float* dv_g2b  = (const float*)d_in[20];  // [1]
  (void)in_sizes; (void)n_in; (void)out_size; (void)ws_size;

  float* out_main = (float*)d_out;                          // [2048,300]
  float* out_k = out_main + (long)ROWS * DD;                // [8,256,256,100]
  float* out_v = out_k + (long)BB * SS * SS * PD;

  // Workspace layout (floats), with reuse:
  float* ws = (float*)d_ws;
  float* P      = ws;                        // 614,400   (dead after A_cat)
  float* A_cat  = ws + 614400;               // 7,987,200 (dead after h)
  float* W_catT = ws + 8601600;              // 1,170,000 (dead after h)
  float* h      = ws + 9771600;              // 614,400
  float* waT    = ws + 10386000;             // 153,600  [768,200]
  float* wbT    = ws + 10539600;             // 230,400  [768,300]
  float* w1T    = ws + 10770000;             // 90,000   [300,300]
  float* w2T    = ws + 10860000;             // 90,000   [300,300]
  float* tfT    = ws + 10950000;             // 614,400  [8][300,256]
  float* mapped_a = A_cat;                   // 1,572,864 (reuse A_cat space)
  float* mapped_b = A_cat + 1572864;         // 1,572,864
  float* scores   = A_cat + 3145728;         // 524,288
  float* out1     = A_cat + 3670016;         // 614,400
  float* x1 = W_catT;                        // reuse W_catT space
  float* x2 = P;                             // reuse P space

  // --- rel_emb broadcast (independent; biggest memory stream) ---
  rel_emb_kernel<<<dim3(ROWS), dim3(256), 0, stream>>>(pe_k, pe_v, out_k, out_v);

  // --- transpose weights to K-major for vectorized B-fragment loads ---
  transpose_kernel<<<dim3((EE * CAH + 255) / 256, 1), dim3(256), 0, stream>>>(
      ca_wa, waT, EE, CAH, 0, 0);
  transpose_kernel<<<dim3((DD * CAH + 255) / 256, 1), dim3(256), 0, stream>>>(
      ca_wb, wbT, DD, CAH, 0, 0);
  transpose_kernel<<<dim3((DD * DD + 255) / 256, 1), dim3(256), 0, stream>>>(
      dv_w1, w1T, DD, DD, 0, 0);
  transpose_kernel<<<dim3((DD * DD + 255) / 256, 1), dim3(256), 0, stream>>>(
      dv_w2, w2T, DD, DD, 0, 0);
  transpose_kernel<<<dim3((SS * DD + 255) / 256, BB), dim3(256), 0, stream>>>(
      tf, tfT, SS, DD, (long)SS * DD, (long)SS * DD);

  // --- RGCN path ---
  prefix_kernel<<<dim3(BB), dim3(320), 0, stream>>>(x, P);
  build_acat_kernel<<<dim3(ROWS), dim3(320), 0, stream>>>(x, P, A_cat);
  {
    const long total = (long)KCAT * DD;
    build_wcatT_kernel<<<dim3((unsigned)((total + 255) / 256)), dim3(256), 0, stream>>>(
        comp, basis, root, W_catT);
  }
  // h = A_cat @ W_catT^T + rgcn_bias   [2048,3900] x [300,3900]^T
  gemm_f32_wmma<0><<<dim3(3, 16, 1), dim3(256), 0, stream>>>(
      A_cat, 0, KCAT, W_catT, 0, KCAT, h, 0, DD, ROWS, DD, KCAT, rg_bias);

  // --- cross attention (overwrites A_cat region; h already produced) ---
  gemm_f32_wmma<0><<<dim3(6, 16, 1), dim3(256), 0, stream>>>(
      emo, 0, EE, waT, 0, EE, mapped_a, 0, CAH, ROWS, CAH, EE, ca_ba);
  gemm_f32_wmma<0><<<dim3(6, 16, 1), dim3(256), 0, stream>>>(
      tf, 0, DD, wbT, 0, DD, mapped_b, 0, CAH, ROWS, CAH, DD, ca_bb);
  // scores[b] = mapped_a[b] @ mapped_b[b]^T  (mapped_b rows are K-major)
  gemm_f32_wmma<0><<<dim3(2, 2, BB), dim3(256), 0, stream>>>(
      mapped_a, (long)SS * CAH, CAH, mapped_b, (long)SS * CAH, CAH,
      scores, (long)SS * SS, SS, SS, SS, CAH, nullptr);
  softmax_col_kernel<<<dim3(SS, BB), dim3(256), 0, stream>>>(scores);
  // out1[b] = A[b] @ tensor_fixed[b]  (B supplied as tfT[b] = [300,256])
  gemm_f32_wmma<0><<<dim3(3, 2, BB), dim3(256), 0, stream>>>(
      scores, (long)SS * SS, SS, tfT, (long)SS * DD, SS,
      out1, (long)SS * DD, DD, SS, DD, SS, nullptr);

  // --- dual view gate ---
  gemm_f32_wmma<1><<<dim3(3, 16, 1), dim3(256), 0, stream>>>(
      h, 0, DD, w1T, 0, DD, x1, 0, DD, ROWS, DD, DD, dv_b1);
  gemm_f32_wmma<1><<<dim3(3, 16, 1), dim3(256), 0, stream>>>(
      out1, 0, DD, w2T, 0, DD, x2, 0, DD, ROWS, DD, DD, dv_b2);
  gate_kernel<<<dim3(ROWS), dim3(256), 0, stream>>>(
      x1, x2, dv_g1w, dv_g1b, dv_g2w, dv_g2b, out_main);
}